// PackedINTAsymmetricWeightsDecompressor_74440373174408
// MI455X (gfx1250) — compile-verified
//
#include <hip/hip_runtime.h>
#include <hip/hip_bf16.h>

// PackedINTAsymmetricWeightsDecompressor for MI455X (gfx1250).
//
// Layout facts (from the reference):
//   weight      : int32[N_PACKED], each element is ONE byte value in [0,256)
//                 holding two uint4 nibbles (low nibble first).
//   scale, zp   : float32[8192*64] (one per group of 128 unpacked values).
//   output      : fp16[8192*8192], out[i] = (fp16)(zp[g] + w[i]*scale[g]).
//
// One "chunk" = 4 consecutive int32 weight elements = 4 packed bytes
//             = 8 unpacked values = 8 fp16 outputs = one b128 load + one
//               b128 store. 16 chunks per quant group, so a chunk never
//               straddles a group boundary.

typedef int       v4i __attribute__((ext_vector_type(4)));
typedef _Float16  v8h __attribute__((ext_vector_type(8)));

__global__ __launch_bounds__(256) void
PackedINTAsymmetricWeightsDecompressor_74440373174408_kernel(
    const int* __restrict__ weight,
    const float* __restrict__ scale,
    const float* __restrict__ zero_point,
    _Float16* __restrict__ out,
    int n_chunks)
{
    const v4i* __restrict__ wp = (const v4i*)weight;
    v8h* __restrict__ op = (v8h*)out;

    const int stride = (int)(gridDim.x * blockDim.x);
    int c = (int)(blockIdx.x * blockDim.x + threadIdx.x);

    for (; c < n_chunks; c += stride) {
        // Pull the next stride's line toward the caches while we convert
        // this one. Speculative prefetch: OOB translation is silently
        // dropped by hardware (CDNA5 ISA 7.3), emits global_prefetch_b8.
        __builtin_prefetch(wp + c + stride, 0, 0);

        // Read-once stream: non-temporal so 128 MiB of weights don't
        // evict the hot scale/zp tables from L2.
        v4i b = __builtin_nontemporal_load(wp + c);

        // 16 chunks per group of 128 values.
        const int g = c >> 4;
        const float s = scale[g];
        const float z = zero_point[g];

        v8h r;
#pragma unroll
        for (int i = 0; i < 4; ++i) {
            const int byte = b[i];                 // value in [0,256)
            const float lo = (float)(byte & 0xF);  // low nibble first
            const float hi = (float)(byte >> 4);   // byte < 256 -> clean high nibble
            r[2 * i + 0] = (_Float16)__builtin_fmaf(lo, s, z);
            r[2 * i + 1] = (_Float16)__builtin_fmaf(hi, s, z);
        }

        // Write-once stream: non-temporal b128 store.
        __builtin_nontemporal_store(r, op + c);
    }
}

extern "C" void kernel_launch(void* const* d_in, const int* in_sizes, int n_in,
                              void* d_out, int out_size, void* d_ws, size_t ws_size,
                              hipStream_t stream) {
    (void)n_in; (void)out_size; (void)d_ws; (void)ws_size;

    const int*   weight     = (const int*)d_in[0];    // int32, one packed byte each
    const float* scale      = (const float*)d_in[1];  // fp32 (8192*64)
    const float* zero_point = (const float*)d_in[2];  // fp32 (8192*64)
    _Float16*    out        = (_Float16*)d_out;       // fp16 output

    const int n_chunks = in_sizes[0] / 4;             // 8,388,608 b128 chunks

    constexpr int kThreads = 256;                     // 8 wave32 per block
    constexpr int kIters   = 4;                       // chunks per thread
    const int blocks = (n_chunks + kThreads * kIters - 1) / (kThreads * kIters);

    PackedINTAsymmetricWeightsDecompressor_74440373174408_kernel
        <<<blocks, kThreads, 0, stream>>>(weight, scale, zero_point, out, n_chunks);
}